// WeightedMeanAggregator_62646392979655
// MI455X (gfx1250) — compile-verified
//
#include <hip/hip_runtime.h>

// Weighted softmax segment aggregation for MI455X (gfx1250, wave32).
//   out[g,h] = sum_i softmax_g(-log_p)_i * x[i,h],  batch sorted.
// Bandwidth-bound: 512MB embeddings @ 23.3TB/s ~= 22us floor.
// Aggregation runs on V_WMMA_F32_16X16X4_F32:
//   block = 16 graphs (M), 8 waves x 16-col tiles (N), K = 4 rows/step.
// Rows are split over grid.y slices (partials in d_ws) so enough loads are
// in flight per SIMD (~14KB needed at 23.3TB/s) to reach the HBM roof.

#define TOTAL_SUBGRAPHS 1048576
#define HIDDEN_DIM 128
#define NUM_GRAPHS 4096
#define GPB 16            // graphs per block = WMMA M
#define THREADS 256       // 8 wave32
#define CHUNK 2048
#define MAX_SPLITS 8

typedef __attribute__((ext_vector_type(2))) float v2f;
typedef __attribute__((ext_vector_type(8))) float v8f;

__device__ __forceinline__ bool finiteBits(float f) {
    return (__float_as_uint(f) & 0x7f800000u) != 0x7f800000u;
}
__device__ __forceinline__ int lowerBound(const int* __restrict__ b, int n, int val) {
    int lo = 0, hi = n;
    while (lo < hi) { int mid = (lo + hi) >> 1; if (b[mid] < val) lo = mid + 1; else hi = mid; }
    return lo;
}

__global__ __launch_bounds__(THREADS)
void wmagg_main(const float* __restrict__ x,
                const int* __restrict__ batch,
                const float* __restrict__ logp,
                float* __restrict__ dst,       // [nsplits][NUM_GRAPHS][HIDDEN_DIM]
                int nsplits) {
    __shared__ int   sbound[GPB + 1];   // per-segment row boundaries
    __shared__ float smax[GPB];
    __shared__ float sinv[GPB];
    __shared__ float2 wseg[CHUNK];      // {normalized weight, seg as float}

    const int tid  = threadIdx.x;
    const int g0   = blockIdx.x * GPB;
    const int wave = tid >> 5;
    const int lane = tid & 31;
    const int half = lane >> 4;         // 0: K rows {0,1}, 1: K rows {2,3}
    const int m    = lane & 15;         // A row M / B col N in tile
    const int n0   = wave * 16;
    const int col  = n0 + m;
    const float fm = (float)m;
    const float NEGINF = -__builtin_inff();

    // ---- segment boundaries via binary search on sorted batch ----
    if (tid <= GPB) sbound[tid] = lowerBound(batch, TOTAL_SUBGRAPHS, g0 + tid);
    __syncthreads();

    // ---- per-segment softmax stats, one wave owns 2 segments ----
    // Shuffle-tree reductions: fixed order -> bit-deterministic, no atomics.
    for (int q = 2 * wave; q < 2 * wave + 2; ++q) {
        const int qs = sbound[q], qe = sbound[q + 1];
        float mx = NEGINF;
        for (int r = qs + lane; r < qe; r += 32) {
            float lp = logp[r];
            mx = fmaxf(mx, finiteBits(lp) ? -lp : NEGINF);
        }
        for (int off = 16; off; off >>= 1) mx = fmaxf(mx, __shfl_down(mx, off, 32));
        mx = __shfl(mx, 0, 32);
        mx = finiteBits(mx) ? mx : 0.0f;        // reference's clamp
        float sm = 0.0f;
        for (int r = qs + lane; r < qe; r += 32) {
            float lp = logp[r];
            sm += finiteBits(lp) ? __expf(-lp - mx) : 0.0f;
        }
        for (int off = 16; off; off >>= 1) sm += __shfl_down(sm, off, 32);
        if (lane == 0) { smax[q] = mx; sinv[q] = 1.0f / fmaxf(sm, 1e-8f); }
    }
    __syncthreads();

    // ---- row slice for this split ----
    const int rstart = sbound[0], rend = sbound[GPB];
    const int lenAll = rend - rstart;
    const int s  = blockIdx.y;
    const int r0 = rstart + (int)((long)lenAll * s / nsplits);
    const int r1 = rstart + (int)((long)lenAll * (s + 1) / nsplits);

    v8f acc = {};   // D: vgpr j -> out[g0 + j + 8*half][col]

    for (int rb = r0; rb < r1; rb += CHUNK) {
        const int clen = min(CHUNK, r1 - rb);

        // stage normalized weights + segment id (one ds_load_b128 per k-block)
        for (int i = tid; i < clen; i += THREADS) {
            int r = rb + i;
            float lp = logp[r];
            int seg = batch[r] - g0;
            float w = finiteBits(lp) ? __expf(-lp - smax[seg]) * sinv[seg] : 0.0f;
            wseg[i] = make_float2(w, (float)seg);
        }
        __syncthreads();

        // hot loop: full K=4 blocks, provably in range -> zero guards
        const int clenFull = clen & ~3;
        const float*  __restrict__ xb  = x + (long)(rb + 2 * half) * HIDDEN_DIM + col;
        const float2* wsg = wseg + 2 * half;
#pragma unroll 8
        for (int kb = 0; kb < clenFull; kb += 4) {
            float2 p0 = wsg[kb];
            float2 p1 = wsg[kb + 1];
            v2f a, b;
            a.x = (p0.y == fm) ? p0.x : 0.0f;      // v_cndmask, no branch
            a.y = (p1.y == fm) ? p1.x : 0.0f;
            b.x = xb[(long)kb * HIDDEN_DIM];        // global_load_b32 pair
            b.y = xb[(long)kb * HIDDEN_DIM + HIDDEN_DIM];
            acc = __builtin_amdgcn_wmma_f32_16x16x4_f32(
                false, a, false, b, (short)0, acc, false, false);
        }
        // masked tail block (clamped indices, weight=0 kills padding rows)
        if (clenFull < clen) {
            int i0 = clenFull + 2 * half, i1 = i0 + 1;
            int j0 = min(i0, clen - 1), j1 = min(i1, clen - 1);
            bool in0 = i0 < clen, in1 = i1 < clen;
            float2 p0 = wseg[j0], p1 = wseg[j1];
            v2f a, b;
            a.x = (in0 & (p0.y == fm)) ? p0.x : 0.0f;
            a.y = (in1 & (p1.y == fm)) ? p1.x : 0.0f;
            b.x = x[(long)(rb + j0) * HIDDEN_DIM + col];
            b.y = x[(long)(rb + j1) * HIDDEN_DIM + col];
            acc = __builtin_amdgcn_wmma_f32_16x16x4_f32(
                false, a, false, b, (short)0, acc, false, false);
        }
        __syncthreads();   // wseg reused next chunk
    }

    // D layout: VGPR j -> M = j (lanes 0-15) / j+8 (lanes 16-31), N = m
    float* outp = dst + (long)blockIdx.y * NUM_GRAPHS * HIDDEN_DIM;
#pragma unroll
    for (int j = 0; j < 8; ++j) {
        int g = g0 + j + 8 * half;
        outp[(long)g * HIDDEN_DIM + col] = acc[j];
    }
}

__global__ __launch_bounds__(256)
void wmagg_reduce(const float* __restrict__ ws, float* __restrict__ out, int nsplits) {
    int i = blockIdx.x * blockDim.x + threadIdx.x;   // over NUM_GRAPHS*HIDDEN_DIM
    float s = 0.0f;
    for (int k = 0; k < nsplits; ++k)
        s += ws[(long)k * NUM_GRAPHS * HIDDEN_DIM + i];
    out[i] = s;
}

extern "C" void kernel_launch(void* const* d_in, const int* in_sizes, int n_in,
                              void* d_out, int out_size, void* d_ws, size_t ws_size,
                              hipStream_t stream) {
    (void)in_sizes; (void)n_in; (void)out_size;
    const float* x     = (const float*)d_in[0];
    const int*   batch = (const int*)d_in[1];
    const float* logp  = (const float*)d_in[2];
    float*       out   = (float*)d_out;

    // largest split count whose partial buffer fits the provided workspace
    int nsplits = 1;
    for (int c = MAX_SPLITS; c >= 2; c >>= 1) {
        size_t need = (size_t)c * NUM_GRAPHS * HIDDEN_DIM * sizeof(float);
        if (ws_size >= need) { nsplits = c; break; }
    }
    float* dst = (nsplits > 1) ? (float*)d_ws : out;

    dim3 grid(NUM_GRAPHS / GPB, nsplits);   // 256 x nsplits blocks
    wmagg_main<<<grid, dim3(THREADS), 0, stream>>>(x, batch, logp, dst, nsplits);
    if (nsplits > 1) {
        wmagg_reduce<<<(NUM_GRAPHS * HIDDEN_DIM) / 256, 256, 0, stream>>>(
            (const float*)d_ws, out, nsplits);
    }
}